// Global_Attention_5111011083039
// MI455X (gfx1250) — compile-verified
//
#include <hip/hip_runtime.h>
#include <hip/hip_bf16.h>
#include <math.h>

#define NUM_SEGMENTS 256
#define HIDDEN 512

typedef __attribute__((ext_vector_type(2))) float v2f;
typedef __attribute__((ext_vector_type(8))) float v8f;

// ---------------------------------------------------------------------------
// Pass 1: gate[i] = x[i,:] . W + b   via V_WMMA_F32_16X16X4_F32
// One wave per 16-row tile of x; K looped in chunks of 4.
// B matrix = W[k..k+3] replicated across all 16 columns, so every column of
// D holds the gate values; we read column 0 (lanes 0 and 16).
// ---------------------------------------------------------------------------
__global__ __launch_bounds__(128)
void gate_wmma_kernel(const float* __restrict__ x,
                      const float* __restrict__ W,
                      const float* __restrict__ bias,
                      float* __restrict__ gate,
                      int n_nodes, int n_tiles)
{
    __shared__ float w_lds[HIDDEN];
    const int tid = threadIdx.x;
    for (int i = tid; i < HIDDEN; i += blockDim.x) w_lds[i] = W[i];
    __syncthreads();

    const int wave = tid >> 5;           // 4 waves / block
    const int lane = tid & 31;
    const long tile = (long)blockIdx.x * 4 + wave;
    if (tile >= n_tiles) return;         // wave-uniform exit (EXEC stays full)

    const long m0   = tile * 16;
    const int  mrow = lane & 15;
    const int  koff = (lane >> 4) << 1;  // lanes 0-15 -> K {0,1}; 16-31 -> K {2,3}

    long m = m0 + mrow;
    if (m >= n_nodes) m = n_nodes - 1;   // clamp tail rows (stores predicated later)
    const float* __restrict__ xrow = x + m * (long)HIDDEN;

    v8f c = {0.f, 0.f, 0.f, 0.f, 0.f, 0.f, 0.f, 0.f};

    #pragma unroll 4
    for (int k0 = 0; k0 < HIDDEN; k0 += 4) {
        const int kb = k0 + koff;
        // A: 16x4 f32 tile, per-lane float2 (8B aligned)
        v2f a;
        a.x = xrow[kb];
        a.y = xrow[kb + 1];
        // B: 4x16 tile, W chunk replicated across all 16 columns
        v2f bv;
        bv.x = w_lds[kb];
        bv.y = w_lds[kb + 1];
        c = __builtin_amdgcn_wmma_f32_16x16x4_f32(
                /*neg_a=*/false, a, /*neg_b=*/false, bv,
                /*c_mod=*/(short)0, c, /*reuse_a=*/false, /*reuse_b=*/false);
    }

    const float bb = bias[0];
    // D layout: VGPR r, lane 0 -> D[M=r][N=0]; lane 16 -> D[M=8+r][N=0]
    if ((lane & 15) == 0) {
        const long base = m0 + (long)((lane >> 4) << 3);
        #pragma unroll
        for (int r = 0; r < 8; ++r) {
            const long mm = base + r;
            if (mm < n_nodes) gate[mm] = c[r] + bb;
        }
    }
}

// ---------------------------------------------------------------------------
// Pass 2: per-segment softmax + weighted pooling.
// batch is sorted -> block s binary-searches its node range; no atomics.
// out[s,h] = (sum_i e_i * x[i,h]) / (sum_i e_i + 1e-16)
// ---------------------------------------------------------------------------
__global__ __launch_bounds__(256)
void segment_pool_kernel(const float* __restrict__ x,
                         const int*   __restrict__ batch,
                         const float* __restrict__ gate,
                         float* __restrict__ out,
                         int n_nodes)
{
    const int s   = blockIdx.x;
    const int tid = threadIdx.x;   // 0..255, owns columns tid and tid+256

    // lower_bound(s) / lower_bound(s+1) over sorted batch
    int lo = 0, hi = n_nodes;
    while (lo < hi) { int mid = (lo + hi) >> 1; if (batch[mid] < s) lo = mid + 1; else hi = mid; }
    const int start = lo;
    hi = n_nodes;
    while (lo < hi) { int mid = (lo + hi) >> 1; if (batch[mid] < s + 1) lo = mid + 1; else hi = mid; }
    const int end = lo;

    __shared__ float red[256];

    // segment max of gate
    float m = -INFINITY;
    for (int i = start + tid; i < end; i += 256) m = fmaxf(m, gate[i]);
    red[tid] = m; __syncthreads();
    for (int off = 128; off > 0; off >>= 1) {
        if (tid < off) red[tid] = fmaxf(red[tid], red[tid + off]);
        __syncthreads();
    }
    const float gmax = red[0];
    __syncthreads();

    // denom = sum exp(gate - gmax)
    float d = 0.f;
    for (int i = start + tid; i < end; i += 256) d += __expf(gate[i] - gmax);
    red[tid] = d; __syncthreads();
    for (int off = 128; off > 0; off >>= 1) {
        if (tid < off) red[tid] += red[tid + off];
        __syncthreads();
    }
    const float denom = red[0] + 1e-16f;

    // weighted accumulation: coalesced 1024B row reads, 2 columns per thread
    float acc0 = 0.f, acc1 = 0.f;
    for (int i = start; i < end; ++i) {
        const float e = __expf(gate[i] - gmax);
        const float* __restrict__ xr = x + (long)i * HIDDEN;
        acc0 += e * xr[tid];
        acc1 += e * xr[tid + 256];
    }
    out[(long)s * HIDDEN + tid]       = acc0 / denom;
    out[(long)s * HIDDEN + tid + 256] = acc1 / denom;
}

extern "C" void kernel_launch(void* const* d_in, const int* in_sizes, int n_in,
                              void* d_out, int out_size, void* d_ws, size_t ws_size,
                              hipStream_t stream) {
    const float* x     = (const float*)d_in[0];
    const int*   batch = (const int*)  d_in[1];
    const float* W     = (const float*)d_in[2];
    const float* bias  = (const float*)d_in[3];
    float*       out   = (float*)d_out;
    float*       gate  = (float*)d_ws;          // N floats of scratch

    const int n_nodes = in_sizes[1];            // 100000
    const int n_tiles = (n_nodes + 15) / 16;    // 16-row WMMA tiles
    const int blocks1 = (n_tiles + 3) / 4;      // 4 waves (tiles) per block

    gate_wmma_kernel<<<blocks1, 128, 0, stream>>>(x, W, bias, gate, n_nodes, n_tiles);
    segment_pool_kernel<<<NUM_SEGMENTS, 256, 0, stream>>>(x, batch, gate, out, n_nodes);
}